// GNNDecoder_59931973648900
// MI455X (gfx1250) — compile-verified
//
#include <hip/hip_runtime.h>
#include <hip/hip_bf16.h>

#define H 128
#define H2 256

typedef __attribute__((ext_vector_type(2))) float v2f;
typedef __attribute__((ext_vector_type(8))) float v8f;
typedef __attribute__((ext_vector_type(4))) unsigned int u32x4;
typedef __attribute__((ext_vector_type(4))) int i32x4;
typedef __attribute__((ext_vector_type(8))) int i32x8;

// ---------------------------------------------------------------------------
// TDM: stage a [rows x cols(dwords)] fp32 tile (row stride = cols) from global
// into LDS at byte offset lds_off, inserting 2 pad dwords every 128 dwords
// (pad_interval code 6, pad_amount code 1) => LDS row pitch = cols + 2.
// Issued by a single wave; completion via TENSORcnt.
// ---------------------------------------------------------------------------
__device__ __forceinline__ void tdm_load_tile_padded(const float* gsrc,
                                                     unsigned lds_off,
                                                     int rows, int cols) {
    unsigned long long ga = (unsigned long long)(uintptr_t)gsrc;
    u32x4 g0;
    g0.x = 1u;                                    // count=1 valid user descriptor
    g0.y = lds_off;                               // lds_addr (bytes), bits 63:32
    g0.z = (unsigned)(ga & 0xFFFFFFFFu);          // global_addr[31:0]
    g0.w = (unsigned)((ga >> 32) & 0x01FFFFFFu)   // global_addr[56:32]
           | (2u << 30);                          // type=2 ("image")
    i32x8 g1;
    g1[0] = (2 << 16)      // data_size = 4 bytes
          | (1 << 20)      // pad_enable
          | (6 << 22)      // pad_interval code 6 = 128 dwords
          | (1 << 25);     // pad_amount  code 1 = 2 dwords
    g1[1] = (cols & 0xFFFF) << 16;                          // tensor_dim0[15:0]
    g1[2] = ((cols >> 16) & 0xFFFF) | ((rows & 0xFFFF) << 16); // dim0[31:16] | dim1[15:0]
    g1[3] = ((rows >> 16) & 0xFFFF) | ((cols & 0xFFFF) << 16); // dim1[31:16] | tile_dim0
    g1[4] = rows & 0xFFFF;                                  // tile_dim1 (tile_dim2 = 0)
    g1[5] = cols;                                           // tensor_dim0_stride[31:0]
    g1[6] = 0;                                              // stride hi | dim1_stride lo
    g1[7] = 0;
    i32x4 z4 = {0, 0, 0, 0};
#if __clang_major__ >= 23
    i32x8 z8 = {0, 0, 0, 0, 0, 0, 0, 0};
    __builtin_amdgcn_tensor_load_to_lds(g0, g1, z4, z4, z8, 0);
#else
    __builtin_amdgcn_tensor_load_to_lds(g0, g1, z4, z4, 0);
#endif
}

// ---------------------------------------------------------------------------
// Kernel 1: h = PReLU(x) @ W_enc^T   (N x 128) = (N x 128)(128 x 128)
// 4 waves/block, one 16-row tile per wave. W_enc staged via TDM into LDS in
// two 64-row halves with pitch 130 (conflict-free B-fragment ds_load_b64).
// f32 WMMA 16x16x4, K-loop of 32 steps.
// ---------------------------------------------------------------------------
#define WPITCH 130
__launch_bounds__(128)
__global__ void enc_gemm_kernel(const float* __restrict__ x,
                                const float* __restrict__ Wenc,
                                const float* __restrict__ prelu_a,
                                float* __restrict__ h, int N) {
    __shared__ float sW[64 * WPITCH];
    const float pa = prelu_a[0];
    const int wave = threadIdx.x >> 5;
    const int lane = threadIdx.x & 31;
    const int r0 = (blockIdx.x * 4 + wave) * 16;

    const int m   = lane & 15;
    const int kko = (lane >> 4) << 1;      // 0 or 2
    const int hih = (lane >> 4) << 3;      // 0 or 8 (row offset of D)
    int arow = r0 + m;
    if (arow >= N) arow = N - 1;           // clamp: no early return (barriers!)

    // Preload all 32 A fragments (whole 16x128 tile) with PReLU applied.
    v2f a[32];
    const float* xr = x + (size_t)arow * H + kko;
#pragma unroll
    for (int ks = 0; ks < 32; ++ks) {
        float u0 = xr[ks * 4];
        float u1 = xr[ks * 4 + 1];
        a[ks].x = u0 > 0.0f ? u0 : pa * u0;
        a[ks].y = u1 > 0.0f ? u1 : pa * u1;
    }

    const unsigned ldsOff = (unsigned)(uintptr_t)(&sW[0]);

    for (int half = 0; half < 2; ++half) {
        if (threadIdx.x == 0) {
            tdm_load_tile_padded(Wenc + (size_t)half * 64 * H, ldsOff, 64, H);
            __builtin_amdgcn_s_wait_tensorcnt(0);
        }
        __syncthreads();                    // LDS tile visible to all waves

        for (int jtl = 0; jtl < 4; ++jtl) {
            const int jt = half * 4 + jtl;
            v8f c = {0.f, 0.f, 0.f, 0.f, 0.f, 0.f, 0.f, 0.f};
            const float* wb = sW + (size_t)(jtl * 16 + m) * WPITCH + kko;
#pragma unroll
            for (int ks = 0; ks < 32; ++ks) {
                v2f b = *(const v2f*)&wb[ks * 4];
                c = __builtin_amdgcn_wmma_f32_16x16x4_f32(false, a[ks], false, b,
                                                          (short)0, c, false, false);
            }
#pragma unroll
            for (int i = 0; i < 8; ++i) {
                int orow = r0 + hih + i;
                if (orow < N) h[(size_t)orow * H + jt * 16 + m] = c[i];
            }
        }
        __syncthreads();                    // before overwriting sW
    }
}

// ---------------------------------------------------------------------------
// Kernel 2: zero masked node rows of h
// ---------------------------------------------------------------------------
__global__ void mask_zero_kernel(float* __restrict__ h,
                                 const int* __restrict__ mask, int M, int N) {
    int id = blockIdx.x * blockDim.x + threadIdx.x;
    int mi = id >> 7;          // 128 cols per row
    int col = id & 127;
    if (mi >= M) return;
    int node = mask[mi];
    if (node >= 0 && node < N) h[(size_t)node * H + col] = 0.0f;
}

// ---------------------------------------------------------------------------
// Kernel 3: aggr[n] = h[n] + emb1[4] + emb2[0]   (self-loop contribution,
// avoids N extra atomic edges and initializes the accumulator)
// ---------------------------------------------------------------------------
__global__ void selfloop_init_kernel(const float* __restrict__ h,
                                     const float* __restrict__ emb1,
                                     const float* __restrict__ emb2,
                                     float* __restrict__ aggr, int N) {
    int id = blockIdx.x * blockDim.x + threadIdx.x;   // N*32 work items
    int node = id >> 5;
    int lane = id & 31;
    if (node >= N) return;
    const float4* hp = (const float4*)(h + (size_t)node * H);
    const float4* e1 = (const float4*)(emb1 + 4 * H);   // bond type 4
    const float4* e2 = (const float4*)(emb2 + 0 * H);   // direction 0
    float4 a = hp[lane], b = e1[lane], c = e2[lane];
    float4 r = {a.x + b.x + c.x, a.y + b.y + c.y, a.z + b.z + c.z, a.w + b.w + c.w};
    ((float4*)(aggr + (size_t)node * H))[lane] = r;
}

// ---------------------------------------------------------------------------
// Kernel 4: edge scatter. Grid-stride, one wave per edge iteration; float4
// per lane (512B rows), 4 non-returning global_atomic_add_f32 per lane into
// L2-resident aggr. Prefetch (global_prefetch_b8) of the NEXT edge's h row
// overlaps the random gather latency. (FP atomics: order nondeterministic.)
// ---------------------------------------------------------------------------
__launch_bounds__(256)
__global__ void edge_scatter_kernel(const float* __restrict__ h,
                                    const int* __restrict__ eidx,
                                    const int* __restrict__ eattr,
                                    const float* __restrict__ emb1,
                                    const float* __restrict__ emb2,
                                    float* __restrict__ aggr, int E,
                                    int numWaves) {
    const int wid  = (blockIdx.x * blockDim.x + threadIdx.x) >> 5;
    const int lane = threadIdx.x & 31;
    for (int e = wid; e < E; e += numWaves) {
        int nxt = e + numWaves;
        if (nxt < E) {
            int ns = eidx[nxt];
            __builtin_prefetch(h + (size_t)ns * H + lane * 4, 0, 3);
        }
        int src = eidx[e];
        int dst = eidx[E + e];
        int t0 = eattr[2 * e];
        int t1 = eattr[2 * e + 1];
        const float4* hp = (const float4*)(h + (size_t)src * H);
        const float4* e1 = (const float4*)(emb1 + (size_t)t0 * H);
        const float4* e2 = (const float4*)(emb2 + (size_t)t1 * H);
        float4 a = hp[lane], b = e1[lane], c = e2[lane];
        float* dp = aggr + (size_t)dst * H + lane * 4;
        atomicAdd(dp + 0, a.x + b.x + c.x);
        atomicAdd(dp + 1, a.y + b.y + c.y);
        atomicAdd(dp + 2, a.z + b.z + c.z);
        atomicAdd(dp + 3, a.w + b.w + c.w);
    }
}

// ---------------------------------------------------------------------------
// Kernel 5: fused GIN MLP: out = relu(aggr @ W1^T + b1) @ W2^T + b2
// One wave per 16-node tile; hidden 16x256 tile round-trips through padded
// LDS (pitch 260 floats => bank-conflict-free D->A fragment transpose).
// Weights read as B-fragments straight from global (L2-hot across blocks).
// ---------------------------------------------------------------------------
#define TPITCH 260
__launch_bounds__(64)
__global__ void mlp_kernel(const float* __restrict__ aggr,
                           const float* __restrict__ W1,
                           const float* __restrict__ b1,
                           const float* __restrict__ W2,
                           const float* __restrict__ b2,
                           float* __restrict__ out, int N) {
    __shared__ float tls[2][16 * TPITCH];
    const int wave = threadIdx.x >> 5;
    const int lane = threadIdx.x & 31;
    const int r0 = (blockIdx.x * 2 + wave) * 16;
    if (r0 >= N) return;                    // no barriers in this kernel

    const int m   = lane & 15;
    const int kko = (lane >> 4) << 1;
    const int hih = (lane >> 4) << 3;
    int arow = r0 + m;
    if (arow >= N) arow = N - 1;
    float* tb = &tls[wave][0];

    // A fragments of aggr tile (16x128), preloaded once.
    v2f a1[32];
    const float* ar = aggr + (size_t)arow * H + kko;
#pragma unroll
    for (int ks = 0; ks < 32; ++ks) {
        a1[ks].x = ar[ks * 4];
        a1[ks].y = ar[ks * 4 + 1];
    }

    // GEMM A: t = relu(aggr @ W1^T + b1), 16 column tiles of 16.
    for (int jt = 0; jt < 16; ++jt) {
        float bias = b1[jt * 16 + m];
        v8f c = {bias, bias, bias, bias, bias, bias, bias, bias};
        const float* wr = W1 + (size_t)(jt * 16 + m) * H + kko;
#pragma unroll
        for (int ks = 0; ks < 32; ++ks) {
            v2f b;
            b.x = wr[ks * 4];
            b.y = wr[ks * 4 + 1];
            c = __builtin_amdgcn_wmma_f32_16x16x4_f32(false, a1[ks], false, b,
                                                      (short)0, c, false, false);
        }
#pragma unroll
        for (int i = 0; i < 8; ++i) {
            float v = c[i];
            tb[(hih + i) * TPITCH + jt * 16 + m] = v > 0.0f ? v : 0.0f;
        }
    }

    // GEMM B: out = t @ W2^T + b2, K = 256 (64 k-steps), A from LDS.
    for (int ot = 0; ot < 8; ++ot) {
        float bias = b2[ot * 16 + m];
        v8f c = {bias, bias, bias, bias, bias, bias, bias, bias};
        const float* wr = W2 + (size_t)(ot * 16 + m) * H2 + kko;
#pragma unroll
        for (int ks = 0; ks < 64; ++ks) {
            v2f a2 = *(const v2f*)&tb[m * TPITCH + ks * 4 + kko];
            v2f b;
            b.x = wr[ks * 4];
            b.y = wr[ks * 4 + 1];
            c = __builtin_amdgcn_wmma_f32_16x16x4_f32(false, a2, false, b,
                                                      (short)0, c, false, false);
        }
#pragma unroll
        for (int i = 0; i < 8; ++i) {
            int orow = r0 + hih + i;
            if (orow < N) out[(size_t)orow * H + ot * 16 + m] = c[i];
        }
    }
}

// ---------------------------------------------------------------------------
// Host launcher. Input order: x, edge_index, edge_attr, mask_node_indices,
// prelu_a, W_enc, W1, b1, W2, b2, emb1, emb2.
// Workspace: h (N*128 f32) | aggr (N*128 f32)  = 51.2 MB.
// ---------------------------------------------------------------------------
extern "C" void kernel_launch(void* const* d_in, const int* in_sizes, int n_in,
                              void* d_out, int out_size, void* d_ws, size_t ws_size,
                              hipStream_t stream) {
    const float* x     = (const float*)d_in[0];
    const int*   eidx  = (const int*)d_in[1];
    const int*   eattr = (const int*)d_in[2];
    const int*   mask  = (const int*)d_in[3];
    const float* pa    = (const float*)d_in[4];
    const float* Wenc  = (const float*)d_in[5];
    const float* W1    = (const float*)d_in[6];
    const float* b1    = (const float*)d_in[7];
    const float* W2    = (const float*)d_in[8];
    const float* b2    = (const float*)d_in[9];
    const float* emb1  = (const float*)d_in[10];
    const float* emb2  = (const float*)d_in[11];
    float* out = (float*)d_out;

    const int N = in_sizes[0] / H;
    const int E = in_sizes[1] / 2;
    const int M = in_sizes[3];

    float* h    = (float*)d_ws;
    float* aggr = h + (size_t)N * H;

    // 1) h = PReLU(x) @ W_enc^T  (TDM-staged weights)
    enc_gemm_kernel<<<dim3((N + 63) / 64), dim3(128), 0, stream>>>(x, Wenc, pa, h, N);
    // 2) zero masked rows
    mask_zero_kernel<<<dim3((M * 128 + 255) / 256), dim3(256), 0, stream>>>(h, mask, M, N);
    // 3) aggr = h + emb1[4] + emb2[0]  (self-loops + init)
    selfloop_init_kernel<<<dim3((N * 32 + 255) / 256), dim3(256), 0, stream>>>(h, emb1, emb2, aggr, N);
    // 4) scatter-add edge messages (grid-stride + prefetch)
    const int blocks = 4096;
    const int numWaves = blocks * (256 / 32);
    edge_scatter_kernel<<<dim3(blocks), dim3(256), 0, stream>>>(h, eidx, eattr, emb1, emb2,
                                                                aggr, E, numWaves);
    // 5) fused MLP -> out
    mlp_kernel<<<dim3((N + 31) / 32), dim3(64), 0, stream>>>(aggr, W1, b1, W2, b2, out, N);
}